// TripletLoss_76373108458148
// MI455X (gfx1250) — compile-verified
//
#include <hip/hip_runtime.h>

typedef __attribute__((ext_vector_type(2))) float v2f;
typedef __attribute__((ext_vector_type(8))) float v8f;

#define NPTS 384
#define DIM  128
#define TRIP_MARGIN 0.2f
#define NTILE (NPTS / 16)   // 24 tiles per dimension

// ---------------------------------------------------------------------------
// Kernel 1: row squared-norms r[i] = sum_k feat[i,k]^2
// ---------------------------------------------------------------------------
__global__ __launch_bounds__(128) void row_norms_kernel(
    const float* __restrict__ feat, float* __restrict__ r) {
  int row = blockIdx.x * blockDim.x + threadIdx.x;
  if (row < NPTS) {
    const float4* f = (const float4*)(feat + row * DIM);
    float s = 0.f;
#pragma unroll
    for (int k = 0; k < DIM / 4; ++k) {
      float4 v = f[k];
      s += v.x * v.x + v.y * v.y + v.z * v.z + v.w * v.w;
    }
    r[row] = s;
  }
}

// ---------------------------------------------------------------------------
// Kernel 2: Gram matrix via V_WMMA_F32_16X16X4_F32, fused into squared
// distances d2[i,j] = max(r[i] + r[j] - 2*<f_i, f_j>, 0).
// One wave32 computes one 16x16 tile; 4 waves per block; 576 tiles total.
// A-matrix 16x4 f32 layout: lane L holds row M = L%16, K-pair (2*(L/16)+{0,1})
// as a float2; B-matrix 4x16 mirrors it (B = feat^T tile, so same loads with
// the column tile's rows). C/D layout: lane L holds N = L%16, M = v + 8*(L/16).
// ---------------------------------------------------------------------------
__global__ __launch_bounds__(128) void gram_d2_kernel(
    const float* __restrict__ feat, const float* __restrict__ r,
    float* __restrict__ d2) {
  int lane = threadIdx.x;                 // 0..31, full wave (EXEC all ones)
  int wave = threadIdx.y;                 // 0..3
  int tile = blockIdx.x * 4 + wave;       // 0..575
  int ti = tile / NTILE;
  int tj = tile % NTILE;
  int i0 = ti * 16;
  int j0 = tj * 16;
  int m    = lane & 15;
  int half = lane >> 4;                   // 0 or 1

  const float* arow = feat + (i0 + m) * DIM;  // A: rows of the i-tile
  const float* brow = feat + (j0 + m) * DIM;  // B: rows of the j-tile (= B cols)

  v8f c = {};
#pragma unroll
  for (int k0 = 0; k0 < DIM; k0 += 4) {
    int kb = k0 + 2 * half;
    v2f a = *(const v2f*)(arow + kb);     // 8B aligned (kb even)
    v2f b = *(const v2f*)(brow + kb);
    // 8 args: (neg_a, A, neg_b, B, c_mod, C, reuse_a, reuse_b)
    c = __builtin_amdgcn_wmma_f32_16x16x4_f32(
        false, a, false, b, (short)0, c, false, false);
  }

  int gj = j0 + m;
  float rj = r[gj];
#pragma unroll
  for (int v = 0; v < 8; ++v) {
    int gi = i0 + v + 8 * half;
    float val = r[gi] + rj - 2.0f * c[v];
    d2[gi * NPTS + gj] = fmaxf(val, 0.0f);
  }
}

// ---------------------------------------------------------------------------
// Kernel 3: per-anchor batch-all triplet partial sums (deterministic: fixed
// index->thread grouping, no float atomics).
// ---------------------------------------------------------------------------
__global__ __launch_bounds__(256) void triplet_rows_kernel(
    const float* __restrict__ d2, const int* __restrict__ y,
    float* __restrict__ tot, float* __restrict__ cnt) {
  __shared__ float sd[NPTS];
  __shared__ int   sy[NPTS];
  __shared__ float red0[256];
  __shared__ float red1[256];

  int a   = blockIdx.x;
  int tid = threadIdx.x;

  for (int j = tid; j < NPTS; j += 256) {
    sd[j] = d2[a * NPTS + j];
    sy[j] = y[j];
  }
  __syncthreads();

  int   ya = sy[a];
  float lt = 0.f;
  float lc = 0.f;
  // Outer loop over candidate positives is uniform across the block;
  // inner loop over negatives is strided by thread id.
  for (int p = 0; p < NPTS; ++p) {
    if (sy[p] == ya && p != a) {
      float dpm = sd[p] + TRIP_MARGIN;
      for (int n = tid; n < NPTS; n += 256) {
        if (sy[n] != ya) {
          lt += fmaxf(dpm - sd[n], 0.0f);
          lc += 1.0f;
        }
      }
    }
  }

  red0[tid] = lt;
  red1[tid] = lc;
  __syncthreads();
#pragma unroll
  for (int s = 128; s > 0; s >>= 1) {
    if (tid < s) {
      red0[tid] += red0[tid + s];
      red1[tid] += red1[tid + s];
    }
    __syncthreads();
  }
  if (tid == 0) {
    tot[a] = red0[0];
    cnt[a] = red1[0];
  }
}

// ---------------------------------------------------------------------------
// Kernel 4: final reduction over anchors, mean = total / count
// ---------------------------------------------------------------------------
__global__ __launch_bounds__(384) void finalize_kernel(
    const float* __restrict__ tot, const float* __restrict__ cnt,
    float* __restrict__ out) {
  __shared__ float r0[NPTS];
  __shared__ float r1[NPTS];
  int t = threadIdx.x;
  r0[t] = tot[t];
  r1[t] = cnt[t];
  __syncthreads();
  if (t < 128) { r0[t] += r0[t + 256]; r1[t] += r1[t + 256]; }
  __syncthreads();
  if (t < 128) { r0[t] += r0[t + 128]; r1[t] += r1[t + 128]; }
  __syncthreads();
#pragma unroll
  for (int s = 64; s > 0; s >>= 1) {
    if (t < s) { r0[t] += r0[t + s]; r1[t] += r1[t + s]; }
    __syncthreads();
  }
  if (t == 0) out[0] = r0[0] / r1[0];
}

// ---------------------------------------------------------------------------
// Launch: feat[384x128] f32, logits (unused), y[384] int
// ws layout (floats): d2[N*N] | r[N] | tot[N] | cnt[N]  (~580 KB)
// ---------------------------------------------------------------------------
extern "C" void kernel_launch(void* const* d_in, const int* in_sizes, int n_in,
                              void* d_out, int out_size, void* d_ws, size_t ws_size,
                              hipStream_t stream) {
  const float* feat = (const float*)d_in[0];
  const int*   y    = (const int*)d_in[2];
  float* ws  = (float*)d_ws;
  float* d2  = ws;
  float* r   = ws + NPTS * NPTS;
  float* tot = ws + NPTS * NPTS + NPTS;
  float* cnt = ws + NPTS * NPTS + 2 * NPTS;
  float* out = (float*)d_out;

  row_norms_kernel<<<3, 128, 0, stream>>>(feat, r);
  gram_d2_kernel<<<dim3((NTILE * NTILE) / 4), dim3(32, 4), 0, stream>>>(feat, r, d2);
  triplet_rows_kernel<<<NPTS, 256, 0, stream>>>(d2, y, tot, cnt);
  finalize_kernel<<<1, NPTS, 0, stream>>>(tot, cnt, out);
}